// GIN_13632226197859
// MI455X (gfx1250) — compile-verified
//
#include <hip/hip_runtime.h>
#include <hip/hip_bf16.h>

typedef __attribute__((ext_vector_type(2))) float v2f;
typedef __attribute__((ext_vector_type(8))) float v8f;

#define N_NODES 50000
#define N_EDGES 800000
#define N_GRAPH 256
#define D_IN    128
#define D_H     64
#define D_OUT   16

// ---------------------------------------------------------------------------
// z = (1 + eps) * x   (elementwise, eps is a device scalar)
// ---------------------------------------------------------------------------
__global__ void gin_init_z(const float* __restrict__ x,
                           const float* __restrict__ eps,
                           float* __restrict__ z, long n) {
  long idx = (long)blockIdx.x * blockDim.x + threadIdx.x;
  if (idx >= n) return;
  z[idx] = (1.0f + eps[0]) * x[idx];
}

// ---------------------------------------------------------------------------
// z[dst[e], :] += x[src[e], :]   one thread per (edge, feature)
// ---------------------------------------------------------------------------
__global__ void gin_edge_scatter(const float* __restrict__ x,
                                 const int* __restrict__ src,
                                 const int* __restrict__ dst,
                                 float* __restrict__ z,
                                 long total, int logD) {
  long idx = (long)blockIdx.x * blockDim.x + threadIdx.x;
  if (idx >= total) return;
  int e = (int)(idx >> logD);
  int f = (int)(idx & ((1 << logD) - 1));
  float v = x[((size_t)src[e] << logD) + f];
  atomicAdd(&z[((size_t)dst[e] << logD) + f], v);
}

// ---------------------------------------------------------------------------
// out[N x 64] = A[N x K] @ W[K x 64] + bias   via V_WMMA_F32_16X16X4_F32.
// One wave per 16-row tile; 4 accumulators cover the 64 output columns.
// A frag: lane holds row (lane&15), K-pair selected by lane>>4.
// C/D frag: VGPR r -> row r (lanes 0-15) / row r+8 (lanes 16-31), col lane&15.
// ---------------------------------------------------------------------------
__global__ void gin_gemm_n64_wmma(const float* __restrict__ A,
                                  const float* __restrict__ W,
                                  const float* __restrict__ bias,
                                  float* __restrict__ out,
                                  int nTiles, int K) {
  int wave = threadIdx.x >> 5;
  int lane = threadIdx.x & 31;
  int tile = blockIdx.x * (blockDim.x >> 5) + wave;
  if (tile >= nTiles) return;

  int m  = lane & 15;          // row within tile (A) / column (B, C/D)
  int kb = (lane >> 4) << 1;   // K-pair base for this half-wave
  const float* arow = A + (size_t)(tile * 16 + m) * K;

  v8f acc0 = {}, acc1 = {}, acc2 = {}, acc3 = {};
  for (int k0 = 0; k0 < K; k0 += 4) {
    v2f a = { arow[k0 + kb], arow[k0 + kb + 1] };
    const float* w0 = W + (size_t)(k0 + kb) * 64;
    const float* w1 = w0 + 64;
    v2f b0 = { w0[m],      w1[m]      };
    v2f b1 = { w0[16 + m], w1[16 + m] };
    v2f b2 = { w0[32 + m], w1[32 + m] };
    v2f b3 = { w0[48 + m], w1[48 + m] };
    acc0 = __builtin_amdgcn_wmma_f32_16x16x4_f32(false, a, false, b0, (short)0, acc0, false, false);
    acc1 = __builtin_amdgcn_wmma_f32_16x16x4_f32(false, a, false, b1, (short)0, acc1, false, false);
    acc2 = __builtin_amdgcn_wmma_f32_16x16x4_f32(false, a, false, b2, (short)0, acc2, false, false);
    acc3 = __builtin_amdgcn_wmma_f32_16x16x4_f32(false, a, false, b3, (short)0, acc3, false, false);
  }

  int col   = lane & 15;
  int rbase = (lane >> 4) * 8;
  float* orow = out + (size_t)(tile * 16 + rbase) * 64;
#pragma unroll
  for (int r = 0; r < 8; ++r) {
    float* o = orow + (size_t)r * 64;
    o[col]      = acc0[r] + bias[col];
    o[16 + col] = acc1[r] + bias[16 + col];
    o[32 + col] = acc2[r] + bias[32 + col];
    o[48 + col] = acc3[r] + bias[48 + col];
  }
}

// ---------------------------------------------------------------------------
// per-channel sum / sum-of-squares over x[N x 64] -> stats[0:64], stats[64:128]
// ---------------------------------------------------------------------------
__global__ void gin_bn_stats(const float* __restrict__ x,
                             float* __restrict__ stats, int n) {
  int c = threadIdx.x & 63;
  int stride = (gridDim.x * blockDim.x) >> 6;
  int r = (blockIdx.x * blockDim.x + threadIdx.x) >> 6;
  float s = 0.0f, s2 = 0.0f;
  for (; r < n; r += stride) {
    float v = x[(size_t)r * 64 + c];
    s += v;
    s2 += v * v;
  }
  atomicAdd(&stats[c], s);
  atomicAdd(&stats[64 + c], s2);
}

// ---------------------------------------------------------------------------
// y = relu(gamma * (x - mean) * rsqrt(var + eps) + beta), biased batch stats
// ---------------------------------------------------------------------------
__global__ void gin_bn_apply_relu(const float* __restrict__ x,
                                  const float* __restrict__ stats,
                                  const float* __restrict__ gamma,
                                  const float* __restrict__ beta,
                                  float* __restrict__ y, int n) {
  long idx = (long)blockIdx.x * blockDim.x + threadIdx.x;
  if (idx >= (long)n * 64) return;
  int c = (int)(idx & 63);
  float invN = 1.0f / (float)n;
  float mean = stats[c] * invN;
  float var  = stats[64 + c] * invN - mean * mean;
  float v = gamma[c] * (x[idx] - mean) * rsqrtf(var + 1e-5f) + beta[c];
  y[idx] = v > 0.0f ? v : 0.0f;
}

// ---------------------------------------------------------------------------
// pooled[node2graph[i], :] += x[i, :]
// ---------------------------------------------------------------------------
__global__ void gin_pool(const float* __restrict__ x,
                         const int* __restrict__ n2g,
                         float* __restrict__ pooled,
                         long total, int logD) {
  long idx = (long)blockIdx.x * blockDim.x + threadIdx.x;
  if (idx >= total) return;
  int i = (int)(idx >> logD);
  int c = (int)(idx & ((1 << logD) - 1));
  atomicAdd(&pooled[((size_t)n2g[i] << logD) + c], x[idx]);
}

// ---------------------------------------------------------------------------
// score[256 x 16] += pooled[256 x K] @ Wp[K x 16] + bias   (WMMA, 16 waves)
// ---------------------------------------------------------------------------
__global__ void gin_head_wmma(const float* __restrict__ pooled,
                              const float* __restrict__ Wp,
                              const float* __restrict__ bias,
                              float* __restrict__ score, int K) {
  int wave = threadIdx.x >> 5;
  int lane = threadIdx.x & 31;
  int tile = blockIdx.x * (blockDim.x >> 5) + wave;
  if (tile >= N_GRAPH / 16) return;

  int m  = lane & 15;
  int kb = (lane >> 4) << 1;
  const float* arow = pooled + (size_t)(tile * 16 + m) * K;

  v8f acc = {};
  for (int k0 = 0; k0 < K; k0 += 4) {
    v2f a = { arow[k0 + kb], arow[k0 + kb + 1] };
    v2f b = { Wp[(size_t)(k0 + kb) * 16 + m], Wp[(size_t)(k0 + kb + 1) * 16 + m] };
    acc = __builtin_amdgcn_wmma_f32_16x16x4_f32(false, a, false, b, (short)0, acc, false, false);
  }

  int col   = lane & 15;
  int rbase = (lane >> 4) * 8;
#pragma unroll
  for (int r = 0; r < 8; ++r) {
    int row = tile * 16 + rbase + r;
    score[(size_t)row * 16 + col] += acc[r] + bias[col];
  }
}

// ---------------------------------------------------------------------------
// Host orchestration. Input flattening (setup_inputs dict order, depth-first):
//   0:h 1:src 2:dst 3:node2graph
//   layer i (i=0..3) at base 4+11i: eps W1 b1 g1 t1 W2 b2 ga ta go to
//   pred head i (i=0..4) at 48+2i: W b
// ---------------------------------------------------------------------------
extern "C" void kernel_launch(void* const* d_in, const int* in_sizes, int n_in,
                              void* d_out, int out_size, void* d_ws, size_t ws_size,
                              hipStream_t stream) {
  const float* h   = (const float*)d_in[0];
  const int*   src = (const int*)d_in[1];
  const int*   dst = (const int*)d_in[2];
  const int*   n2g = (const int*)d_in[3];

  float* ws     = (float*)d_ws;
  float* z      = ws;                          // N x 128 (aggregation buffer)
  float* u      = z  + (size_t)N_NODES * D_IN; // N x 64 (GEMM output)
  float* tA     = u  + (size_t)N_NODES * D_H;  // N x 64 (post-BN temp)
  float* tB     = tA + (size_t)N_NODES * D_H;  // N x 64 (layer output / x)
  float* stats  = tB + (size_t)N_NODES * D_H;  // 128 (sum, sumsq)
  float* pooled = stats + 128;                 // 256 x 128

  const int nTiles = N_NODES / 16;             // 3125 exactly
  const unsigned gemmBlocks = (unsigned)((nTiles + 7) / 8);

  hipMemsetAsync(d_out, 0, (size_t)N_GRAPH * D_OUT * sizeof(float), stream);

  // --- head 0 on raw input h ---
  hipMemsetAsync(pooled, 0, (size_t)N_GRAPH * D_IN * sizeof(float), stream);
  {
    long total = (long)N_NODES * D_IN;
    gin_pool<<<(unsigned)((total + 255) / 256), 256, 0, stream>>>(h, n2g, pooled, total, 7);
  }
  gin_head_wmma<<<2, 256, 0, stream>>>(pooled, (const float*)d_in[48],
                                       (const float*)d_in[49], (float*)d_out, D_IN);

  const float* xcur = h;
  int D = D_IN, logD = 7;

  for (int i = 0; i < 4; ++i) {
    int base = 4 + 11 * i;
    const float* eps = (const float*)d_in[base + 0];
    const float* W1  = (const float*)d_in[base + 1];
    const float* b1  = (const float*)d_in[base + 2];
    const float* g1  = (const float*)d_in[base + 3];
    const float* t1  = (const float*)d_in[base + 4];
    const float* W2  = (const float*)d_in[base + 5];
    const float* b2  = (const float*)d_in[base + 6];
    const float* ga  = (const float*)d_in[base + 7];
    const float* ta  = (const float*)d_in[base + 8];
    const float* go  = (const float*)d_in[base + 9];
    const float* to  = (const float*)d_in[base + 10];

    // z = (1+eps)*x + sum_{j->i} x_j
    long nzd = (long)N_NODES * D;
    gin_init_z<<<(unsigned)((nzd + 255) / 256), 256, 0, stream>>>(xcur, eps, z, nzd);
    long ted = (long)N_EDGES * D;
    gin_edge_scatter<<<(unsigned)((ted + 255) / 256), 256, 0, stream>>>(xcur, src, dst, z, ted, logD);

    // u = z @ W1 + b1 ; tA = relu(bn(u; g1,t1))
    gin_gemm_n64_wmma<<<gemmBlocks, 256, 0, stream>>>(z, W1, b1, u, nTiles, D);
    hipMemsetAsync(stats, 0, 128 * sizeof(float), stream);
    gin_bn_stats<<<256, 256, 0, stream>>>(u, stats, N_NODES);
    gin_bn_apply_relu<<<(unsigned)(((long)N_NODES * 64 + 255) / 256), 256, 0, stream>>>(
        u, stats, g1, t1, tA, N_NODES);

    // u = tA @ W2 + b2
    gin_gemm_n64_wmma<<<gemmBlocks, 256, 0, stream>>>(tA, W2, b2, u, nTiles, D_H);

    // tA = relu(bn(u; ga,ta))
    hipMemsetAsync(stats, 0, 128 * sizeof(float), stream);
    gin_bn_stats<<<256, 256, 0, stream>>>(u, stats, N_NODES);
    gin_bn_apply_relu<<<(unsigned)(((long)N_NODES * 64 + 255) / 256), 256, 0, stream>>>(
        u, stats, ga, ta, tA, N_NODES);

    // tB = relu(bn(tA; go,to))   -> next layer's x
    hipMemsetAsync(stats, 0, 128 * sizeof(float), stream);
    gin_bn_stats<<<256, 256, 0, stream>>>(tA, stats, N_NODES);
    gin_bn_apply_relu<<<(unsigned)(((long)N_NODES * 64 + 255) / 256), 256, 0, stream>>>(
        tA, stats, go, to, tB, N_NODES);

    // pooled head for this layer's representation
    hipMemsetAsync(pooled, 0, (size_t)N_GRAPH * D_H * sizeof(float), stream);
    long tnd = (long)N_NODES * D_H;
    gin_pool<<<(unsigned)((tnd + 255) / 256), 256, 0, stream>>>(tB, n2g, pooled, tnd, 6);
    int pbase = 48 + 2 * (i + 1);
    gin_head_wmma<<<2, 256, 0, stream>>>(pooled, (const float*)d_in[pbase],
                                         (const float*)d_in[pbase + 1], (float*)d_out, D_H);

    xcur = tB;
    D = D_H;
    logD = 6;
  }
}